// pytorch_ffa_19052474925168
// MI455X (gfx1250) — compile-verified
//
#include <hip/hip_runtime.h>
#include <hip/hip_bf16.h>
#include <math.h>

#define T_LEN   2097152     // samples per (b,c)
#define NBC     8           // B*C
#define TARGET  16384       // padded row count (power of 2) for every bins value
#define TILE    128         // rows per LDS tile; also threads per FFA block
#define COLPAD  133         // LDS column stride: odd -> conflict-free row-major access
#define YSTRIDE 132         // global Y column stride
#define NBINS   4           // bins = 128..131
#define PER_OFF 65536       // number of period values at the start of d_out

typedef float v2f __attribute__((ext_vector_type(2)));
typedef float v8f __attribute__((ext_vector_type(8)));

// ---------------------------------------------------------------------------
// K1: partial (sum, sumsq) per 32768-element chunk using WMMA ones-reduction.
// D = A x Ones + C accumulates row-sums of A; summing the accumulator over
// lanes/regs yields the total of all 64 elements fed per WMMA issue.
// ---------------------------------------------------------------------------
__global__ __launch_bounds__(256) void k_reduce(const float* __restrict__ x,
                                                float2* __restrict__ partials) {
  const int bc   = blockIdx.y;
  const int blk  = blockIdx.x;              // 64 blocks per (b,c)
  const int wave = threadIdx.x >> 5;
  const int lane = threadIdx.x & 31;
  const size_t base = (size_t)bc * T_LEN + (size_t)blk * 32768 + (size_t)wave * 4096;
  const float2* p2 = (const float2*)(x + base);

  v8f cs = {};
  v8f cq = {};
  v2f ones; ones[0] = 1.0f; ones[1] = 1.0f;

  for (int it = 0; it < 64; ++it) {
    float2 d = p2[it * 32 + lane];
    __builtin_prefetch((const void*)(p2 + it * 32 + lane + 256), 0, 0);
    v2f a;  a[0]  = d.x;        a[1]  = d.y;
    v2f a2; a2[0] = d.x * d.x;  a2[1] = d.y * d.y;
    cs = __builtin_amdgcn_wmma_f32_16x16x4_f32(false, a,  false, ones, (short)0, cs, false, false);
    cq = __builtin_amdgcn_wmma_f32_16x16x4_f32(false, a2, false, ones, (short)0, cq, false, false);
  }
  float s = cs[0]+cs[1]+cs[2]+cs[3]+cs[4]+cs[5]+cs[6]+cs[7];
  float q = cq[0]+cq[1]+cq[2]+cq[3]+cq[4]+cq[5]+cq[6]+cq[7];
  // lanes 0-15 hold rows 0-7 totals, lanes 16-31 rows 8-15: combine halves.
  s += __shfl_xor(s, 16);
  q += __shfl_xor(q, 16);

  __shared__ float sh_s[8], sh_q[8];
  if (lane == 0) { sh_s[wave] = s; sh_q[wave] = q; }
  __syncthreads();
  if (threadIdx.x == 0) {
    float S = 0.f, Q = 0.f;
    for (int w = 0; w < 8; ++w) { S += sh_s[w]; Q += sh_q[w]; }
    partials[bc * 64 + blk] = make_float2(S, Q);
  }
}

// ---------------------------------------------------------------------------
// K2: finalize 1/std per (b,c)  (ddof=1)
// ---------------------------------------------------------------------------
__global__ void k_finalize_std(const float2* __restrict__ partials,
                               float* __restrict__ invStd) {
  const int bc = threadIdx.x;
  if (bc < NBC) {
    float S = 0.f, Q = 0.f;
    for (int i = 0; i < 64; ++i) { float2 v = partials[bc * 64 + i]; S += v.x; Q += v.y; }
    const double N = (double)T_LEN;
    const double var = ((double)Q - (double)S * (double)S / N) / (N - 1.0);
    invStd[bc] = (float)(1.0 / sqrt(var));
  }
}

// ---------------------------------------------------------------------------
// K3: periods = concat of np.linspace(tsamp*bins, tsamp*(bins+1), 16384)
// ---------------------------------------------------------------------------
__global__ void k_periods(float* __restrict__ out) {
  const int idx  = blockIdx.x * blockDim.x + threadIdx.x;   // 65536 threads
  const int bi   = idx >> 14;
  const int k    = idx & (TARGET - 1);
  const int bins = 128 + bi;
  const double tsamp = 6.4e-05;
  const double p0 = tsamp * (double)bins;
  const double step = tsamp / (double)(TARGET - 1);
  const double v = (k == TARGET - 1) ? tsamp * (double)(bins + 1)
                                     : p0 + (double)k * step;
  out[idx] = (float)v;
}

// ---------------------------------------------------------------------------
// 7 FFA butterfly stages on a 128-row x ncol tile held in LDS (ping-pong).
// Stage s, group g=2^s: out[i] = in[iA] + rot(in[iB], shiftBase*g/2+(il+1)/2)
// shiftBase = 0 for level-1 (stages 1..7), = u for level-2 (stages 8..14).
// 128 threads; returns pointer to the buffer holding the result.
// ---------------------------------------------------------------------------
__device__ __forceinline__ float* ffa_stages7(float* b0, float* b1,
                                              int ncol, int shiftBase) {
  const int tid = threadIdx.x;
  float* src = b0;
  float* dst = b1;
  for (int s = 1; s <= 7; ++s) {
    const int g = 1 << s, h = g >> 1;
    const int cmod = (shiftBase * h) % ncol;
    for (int i = 0; i < TILE; ++i) {
      const int il = i & (g - 1);
      const int iA = (i - il) + (il >> 1);
      const int iB = iA + h;
      int sh = cmod + ((il + 1) >> 1);
      if (sh >= ncol) sh -= ncol;
      const float* rA = src + iA * COLPAD;
      const float* rB = src + iB * COLPAD;
      float* rD = dst + i * COLPAD;
      int j = tid;                       // tid < 128 <= ncol always
      int js = j - sh; if (js < 0) js += ncol;
      rD[j] = rA[j] + rB[js];
      j = tid + TILE;
      if (j < ncol) {
        js = j - sh; if (js < 0) js += ncol;
        rD[j] = rA[j] + rB[js];
      }
    }
    __syncthreads();
    float* t = src; src = dst; dst = t;
  }
  return src;   // after 7 swaps this is b1
}

// ---------------------------------------------------------------------------
// K4 (level 1): load a contiguous 128-row tile of the folded input (zero-pad
// rows >= rowsReal), run stages 1..7 in LDS, store transposed (u-major) to Y:
//   Y[bc][u][q][col]   (global stride YSTRIDE)
// ---------------------------------------------------------------------------
__global__ __launch_bounds__(128) void k_ffa_local(const float* __restrict__ x,
                                                   float* __restrict__ Y,
                                                   int bins, int rowsReal) {
  extern __shared__ float smem[];
  float* b0 = smem;
  float* b1 = smem + TILE * COLPAD;
  const int q  = blockIdx.x;            // tile index 0..127
  const int bc = blockIdx.y;
  const int tid = threadIdx.x;
  const float* xb = x + (size_t)bc * T_LEN;

  for (int u = 0; u < TILE; ++u) {
    const int r = q * TILE + u;
    const bool live = (r < rowsReal);
    const size_t rb = (size_t)r * bins;
    b0[u * COLPAD + tid] = live ? xb[rb + tid] : 0.0f;
    const int j2 = tid + TILE;
    if (j2 < bins) b0[u * COLPAD + j2] = live ? xb[rb + j2] : 0.0f;
  }
  __syncthreads();

  float* res = ffa_stages7(b0, b1, bins, /*shiftBase=*/0);

  for (int u = 0; u < TILE; ++u) {
    float* dstRow = Y + ((size_t)(bc * TILE + u) * TILE + q) * YSTRIDE;
    dstRow[tid] = res[u * COLPAD + tid];
    const int j2 = tid + TILE;
    if (j2 < bins) dstRow[j2] = res[u * COLPAD + j2];
  }
}

// ---------------------------------------------------------------------------
// K5 (level 2 + SNR): block (bc, u) loads the 128 stage-7 rows with intra-tile
// index u (contiguous in Y), runs stages 8..14 as a 128-row FFA with
// shiftBase=u, then computes SNR = (max - median) * invStd * 1/sqrt(rowsReal)
// per profile. Median is the exact k-th smallest via MSB-first radix select on
// order-preserving keys (bitwise identical to sorting), with early exit once
// the candidate set collapses to one element.
// ---------------------------------------------------------------------------
__global__ __launch_bounds__(128) void k_ffa_global_snr(const float* __restrict__ Y,
                                                        float* __restrict__ out,
                                                        const float* __restrict__ invStd,
                                                        int bins, float scale, int binsIdx) {
  extern __shared__ float smem[];
  float* b0 = smem;
  float* b1 = smem + TILE * COLPAD;
  const int u  = blockIdx.x;
  const int bc = blockIdx.y;
  const int tid = threadIdx.x;

  const float* Yb = Y + (size_t)(bc * TILE + u) * TILE * YSTRIDE;
  for (int p = 0; p < TILE; ++p) {
    const float* srcRow = Yb + (size_t)p * YSTRIDE;
    b0[p * COLPAD + tid] = srcRow[tid];
    const int j2 = tid + TILE;
    if (j2 < bins) b0[p * COLPAD + j2] = srcRow[j2];
  }
  __syncthreads();

  float* res = ffa_stages7(b0, b1, bins, /*shiftBase=*/u);

  // Per-thread profile: row d = tid (final row index = u*128 + tid)
  float* row = res + tid * COLPAD;
  unsigned* urow = (unsigned*)row;

  // Single sweep: running max + in-place transform to monotonic unsigned keys.
  float mx = row[0];
  {
    unsigned uv = __float_as_uint(row[0]);
    urow[0] = (uv & 0x80000000u) ? ~uv : (uv | 0x80000000u);
  }
  for (int j = 1; j < bins; ++j) {
    const float f = row[j];
    mx = fmaxf(mx, f);
    const unsigned uv = __float_as_uint(f);
    urow[j] = (uv & 0x80000000u) ? ~uv : (uv | 0x80000000u);
  }

  int k = (bins - 1) >> 1;                 // 0-indexed median position
  int n = bins;                            // current candidate count
  unsigned prefix = 0u, mask = 0u;
  unsigned medKey = 0u;
  for (int b = 31; b >= 0; --b) {
    const unsigned m = mask | (1u << b);
    int cnt = 0;
    for (int j = 0; j < bins; ++j) cnt += ((urow[j] & m) == prefix) ? 1 : 0;
    if (k < cnt) { mask = m; n = cnt; }
    else         { k -= cnt; mask = m; prefix |= (1u << b); n -= cnt; }
    if (n == 1) {
      // Unique candidate: fetch its full key directly.
      for (int j = 0; j < bins; ++j) {
        if ((urow[j] & mask) == prefix) { medKey = urow[j]; break; }
      }
      break;
    }
    medKey = prefix;                       // valid when loop runs to completion
  }
  const float med = (medKey & 0x80000000u) ? __uint_as_float(medKey & 0x7FFFFFFFu)
                                           : __uint_as_float(~medKey);

  const float snr = (mx - med) * invStd[bc] * scale;
  out[PER_OFF + (size_t)bc * (NBINS * TARGET) + (size_t)binsIdx * TARGET
      + u * TILE + tid] = snr;
}

// ---------------------------------------------------------------------------
extern "C" void kernel_launch(void* const* d_in, const int* in_sizes, int n_in,
                              void* d_out, int out_size, void* d_ws, size_t ws_size,
                              hipStream_t stream) {
  (void)in_sizes; (void)n_in; (void)out_size; (void)ws_size;
  const float* x = (const float*)d_in[0];
  float* out = (float*)d_out;
  char* ws = (char*)d_ws;

  float2* partials = (float2*)ws;                 // 512 * 8B
  float*  invStd   = (float*)(ws + 4096);         // 8 floats
  float*  Y        = (float*)(ws + 8192);         // 8*128*128*132*4B ~ 66 MB (reused per bins)

  const size_t ldsBytes = (size_t)2 * TILE * COLPAD * sizeof(float);  // 136192 B

  k_reduce<<<dim3(64, NBC), 256, 0, stream>>>(x, partials);
  k_finalize_std<<<1, 32, 0, stream>>>(partials, invStd);
  k_periods<<<PER_OFF / 256, 256, 0, stream>>>(out);

  for (int bi = 0; bi < NBINS; ++bi) {
    const int bins = 128 + bi;
    const int rowsReal = T_LEN / bins;            // 16384, 16256, 16131, 16008
    const float scale = 1.0f / sqrtf((float)rowsReal);   // 1/sqrt(target - added)
    k_ffa_local<<<dim3(TILE, NBC), TILE, ldsBytes, stream>>>(x, Y, bins, rowsReal);
    k_ffa_global_snr<<<dim3(TILE, NBC), TILE, ldsBytes, stream>>>(Y, out, invStd, bins, scale, bi);
  }
}